// Loss_SSIM_68582037782965
// MI455X (gfx1250) — compile-verified
//
#include <hip/hip_runtime.h>
#include <math.h>

typedef float v2f __attribute__((ext_vector_type(2)));
typedef float v8f __attribute__((ext_vector_type(8)));

#define IMG_H 512
#define IMG_W 512
#define NIMG  32
#define NPIX  (NIMG * IMG_H * IMG_W)

#define SSIM_C1 1.0e-4f
#define SSIM_C2 9.0e-4f

#define XS 29     // LDS input-tile row stride (floats), 28 cols used
#define HS 20     // LDS H-staging row stride (floats), avoids D-write bank conflicts
#define WAVES 4   // waves (16x16 tiles) per workgroup

// Intra-wave LDS producer->consumer sync (wave32: lanes are lockstep; only DS
// completion + scheduling order is needed, no workgroup barrier).
__device__ __forceinline__ void lds_sync() {
    __builtin_amdgcn_wave_barrier();
    asm volatile("s_wait_dscnt 0" ::: "memory");
    __builtin_amdgcn_wave_barrier();
}

__device__ __forceinline__ v8f zero8() {
    v8f z;
#pragma unroll
    for (int i = 0; i < 8; ++i) z[i] = 0.0f;
    return z;
}

// Separable 11-tap Gaussian conv producing one 16x16 output tile.
// Horizontal: H(32x16) = X(32x28) * Gh(28x16)  [two 16-row WMMA passes, K=28 in 7 chunks]
// Vertical:   D(16x16) = Gv(16x28) * H(28x16)  [7 chunks]
// wc[t] holds the banded Gaussian chunk; identical per-lane values serve as the
// horizontal B operand and the vertical A operand (band symmetry + ISA layouts).
__device__ __forceinline__ v8f conv_tile(const float* t1, const float* t2, bool prod,
                                         float* hb, int m, int hi, const v2f* wc)
{
#pragma unroll
    for (int p = 0; p < 2; ++p) {
        v8f acc = zero8();
        const int row = p * 16 + m;     // A-matrix row for this lane
#pragma unroll
        for (int t = 0; t < 7; ++t) {
            const int j0 = 4 * t + 2 * hi;  // logical K index of this lane's pair
            float a0 = t1[row * XS + j0];
            float a1 = t1[row * XS + j0 + 1];
            if (prod) {
                a0 *= t2[row * XS + j0];
                a1 *= t2[row * XS + j0 + 1];
            }
            v2f a; a.x = a0; a.y = a1;
            acc = __builtin_amdgcn_wmma_f32_16x16x4_f32(
                false, a, false, wc[t], (short)0, acc, false, false);
        }
        // D layout: VGPR v -> row (v + 8*hi), col = lane&15
#pragma unroll
        for (int v = 0; v < 8; ++v)
            hb[(p * 16 + v + 8 * hi) * HS + m] = acc[v];
    }
    lds_sync();

    v8f accv = zero8();
#pragma unroll
    for (int t = 0; t < 7; ++t) {
        const int j0 = 4 * t + 2 * hi;
        v2f b;
        b.x = hb[j0 * HS + m];
        b.y = hb[(j0 + 1) * HS + m];
        accv = __builtin_amdgcn_wmma_f32_16x16x4_f32(
            false, wc[t], false, b, (short)0, accv, false, false);
    }
    lds_sync();   // protect hb WAR before next quantity's horizontal writes
    return accv;
}

__global__ __launch_bounds__(WAVES * 32)
void ssim_kernel(const float* __restrict__ ir, const float* __restrict__ vi,
                 const float* __restrict__ fu, double* __restrict__ acc)
{
    __shared__ float s_x[WAVES][3][32 * XS];
    __shared__ float s_h[WAVES][32 * HS];

    const int wv   = threadIdx.x >> 5;
    const int lane = threadIdx.x & 31;
    const int gid  = blockIdx.x * WAVES + wv;
    const int NT   = NIMG * (IMG_H / 16) * (IMG_W / 16);   // 32768 tiles
    if (gid >= NT) return;

    const int img = gid >> 10;          // 1024 tiles per image
    const int tin = gid & 1023;
    const int r0  = ((tin >> 5) << 4) - 5;   // tile origin incl. -5 halo
    const int c0  = ((tin & 31) << 4) - 5;
    const size_t ib = (size_t)img * (IMG_H * IMG_W);

    float* xIR = &s_x[wv][0][0];
    float* xVI = &s_x[wv][1][0];
    float* xFU = &s_x[wv][2][0];
    float* hb  = &s_h[wv][0];

    // Load 32x28 zero-padded input tile for all three images (per-wave).
    for (int idx = lane; idx < 32 * 28; idx += 32) {
        const int r  = idx / 28;
        const int c  = idx - r * 28;
        const int gr = r0 + r, gc = c0 + c;
        const bool in = (gr >= 0) & (gr < IMG_H) & (gc >= 0) & (gc < IMG_W);
        const size_t off = ib + (size_t)gr * IMG_W + gc;
        xIR[r * XS + c] = in ? ir[off] : 0.0f;
        xVI[r * XS + c] = in ? vi[off] : 0.0f;
        xFU[r * XS + c] = in ? fu[off] : 0.0f;
    }
    lds_sync();

    // Normalized 11-tap Gaussian, sigma = 1.5
    const float G11[11] = {0.0010283f, 0.0075981f, 0.0360011f, 0.1093605f, 0.2130059f,
                           0.2660122f, 0.2130059f, 0.1093605f, 0.0360011f, 0.0075981f,
                           0.0010283f};
    const int m  = lane & 15;
    const int hi = lane >> 4;
    v2f wc[7];
#pragma unroll
    for (int t = 0; t < 7; ++t) {
        const int k0 = 4 * t + 2 * hi;
        const int d0 = k0 - m, d1 = k0 + 1 - m;
        wc[t].x = (d0 >= 0 && d0 <= 10) ? G11[d0] : 0.0f;
        wc[t].y = (d1 >= 0 && d1 <= 10) ? G11[d1] : 0.0f;
    }

    // Shared between both SSIM pairs
    v8f mu2 = conv_tile(xFU, xFU, false, hb, m, hi, wc);   // E[fuse]
    v8f w22 = conv_tile(xFU, xFU, true,  hb, m, hi, wc);   // E[fuse^2]

    float ssum[2];
#pragma unroll
    for (int pr = 0; pr < 2; ++pr) {
        const float* x1 = pr ? xVI : xIR;
        v8f mu1 = conv_tile(x1, x1,  false, hb, m, hi, wc);   // E[x]
        v8f w11 = conv_tile(x1, x1,  true,  hb, m, hi, wc);   // E[x^2]
        v8f w12 = conv_tile(x1, xFU, true,  hb, m, hi, wc);   // E[x*fuse]
        float s = 0.0f;
#pragma unroll
        for (int v = 0; v < 8; ++v) {
            const float m1  = mu1[v], m2 = mu2[v];
            const float m1s = m1 * m1, m2s = m2 * m2, m12 = m1 * m2;
            const float s1  = w11[v] - m1s;
            const float s2  = w22[v] - m2s;
            const float s12 = w12[v] - m12;
            s += ((2.0f * m12 + SSIM_C1) * (2.0f * s12 + SSIM_C2)) /
                 ((m1s + m2s + SSIM_C1) * (s1 + s2 + SSIM_C2));
        }
        ssum[pr] = s;
    }

#pragma unroll
    for (int off = 16; off > 0; off >>= 1) {
        ssum[0] += __shfl_xor(ssum[0], off);
        ssum[1] += __shfl_xor(ssum[1], off);
    }
    if (lane == 0) {
        atomicAdd(&acc[2], (double)ssum[0]);   // sum of ssim_map(ir, fuse)
        atomicAdd(&acc[3], (double)ssum[1]);   // sum of ssim_map(vi, fuse)
    }
}

__global__ __launch_bounds__(256)
void sobel_kernel(const float* __restrict__ ir, const float* __restrict__ vi,
                  double* __restrict__ acc)
{
    float sir = 0.0f, svi = 0.0f;
    const int stride = gridDim.x * blockDim.x;
    for (int p = blockIdx.x * blockDim.x + threadIdx.x; p < NPIX; p += stride) {
        const int c = p & (IMG_W - 1);
        const int r = (p >> 9) & (IMG_H - 1);
        const int ibase = (p >> 18) << 18;
        float gxi = 0, gyi = 0, gxv = 0, gyv = 0;
#pragma unroll
        for (int dy = -1; dy <= 1; ++dy) {
#pragma unroll
            for (int dx = -1; dx <= 1; ++dx) {
                const int rr = r + dy, cc = c + dx;
                if (rr < 0 || rr >= IMG_H || cc < 0 || cc >= IMG_W) continue;
                const float wx = (float)dx * ((dy == 0) ? 2.0f : 1.0f);
                const float wy = (float)(-dy) * ((dx == 0) ? 2.0f : 1.0f);
                const size_t off = (size_t)ibase + (size_t)rr * IMG_W + cc;
                const float a = ir[off], b = vi[off];
                gxi += wx * a; gyi += wy * a;
                gxv += wx * b; gyv += wy * b;
            }
        }
        sir += fabsf(gxi) + fabsf(gyi);
        svi += fabsf(gxv) + fabsf(gyv);
    }
#pragma unroll
    for (int off = 16; off > 0; off >>= 1) {
        sir += __shfl_xor(sir, off);
        svi += __shfl_xor(svi, off);
    }
    if ((threadIdx.x & 31) == 0) {
        atomicAdd(&acc[0], (double)sir);
        atomicAdd(&acc[1], (double)svi);
    }
}

__global__ void init_kernel(double* acc) {
    if (threadIdx.x < 4 && blockIdx.x == 0) acc[threadIdx.x] = 0.0;
}

__global__ void final_kernel(const double* __restrict__ acc, float* __restrict__ out) {
    if (threadIdx.x == 0 && blockIdx.x == 0) {
        const double gi = acc[0], gv = acc[1];
        const double wi = gi / (gi + gv);
        const double wv = gv / (gi + gv);
        const double n  = (double)NPIX;
        out[0] = (float)(wi * (acc[2] / n) + wv * (acc[3] / n));
    }
}

extern "C" void kernel_launch(void* const* d_in, const int* in_sizes, int n_in,
                              void* d_out, int out_size, void* d_ws, size_t ws_size,
                              hipStream_t stream) {
    (void)in_sizes; (void)n_in; (void)out_size; (void)ws_size;
    const float* ir = (const float*)d_in[0];
    const float* vi = (const float*)d_in[1];
    const float* fu = (const float*)d_in[2];
    double* acc = (double*)d_ws;
    float* out = (float*)d_out;

    init_kernel<<<1, 32, 0, stream>>>(acc);
    sobel_kernel<<<2048, 256, 0, stream>>>(ir, vi, acc);

    const int ntiles = NIMG * (IMG_H / 16) * (IMG_W / 16);   // 32768
    const int nblocks = (ntiles + WAVES - 1) / WAVES;        // 8192
    ssim_kernel<<<nblocks, WAVES * 32, 0, stream>>>(ir, vi, fu, acc);

    final_kernel<<<1, 32, 0, stream>>>(acc, out);
}